// TFModel_23957327577876
// MI455X (gfx1250) — compile-verified
//
#include <hip/hip_runtime.h>
#include <hip/hip_bf16.h>
#include <math.h>

// ---------------------------------------------------------------------------
// Enformer-style forward pass for gfx1250 (MI455X).
// All matmuls via V_WMMA_F32_16X16X4_F32 (fp32 in / fp32 acc, wave32).
// Each wave computes a 32x32 output tile (2x2 grid of 16x16 WMMA tiles) so
// every operand fragment feeds two wmma issues.
// ---------------------------------------------------------------------------

typedef float v2f __attribute__((ext_vector_type(2)));
typedef float v8f __attribute__((ext_vector_type(8)));

#define DEV __device__ __forceinline__

DEV float gelu_(float x) { return x * (1.f / (1.f + expf(-1.702f * x))); }

DEV v8f wmma4_(v2f a, v2f b, v8f c)
{
    return __builtin_amdgcn_wmma_f32_16x16x4_f32(false, a, false, b,
                                                 (short)0, c, false, false);
}

// ======================= WMMA conv/GEMM (im2row gather) ====================
// C[L x Cout] = act( conv(transform(X[L x Cin]), W[ks*Cin x Cout]) + bias + res )
// If bng != null, input transform = gelu(bn(x)) fused into the A-gather.
__global__ __launch_bounds__(128)
void k_conv_wmma(
    const float* __restrict__ X, int L, int Cin,
    const float* __restrict__ W, int Cout,
    const float* __restrict__ bias, const float* __restrict__ res,
    float* __restrict__ C, int ldc,
    int ks, int pad, int act,
    const float* __restrict__ bng, const float* __restrict__ bnb,
    const float* __restrict__ bnm, const float* __restrict__ bnv)
{
    const int Nt = (Cout + 31) >> 5;
    const int Mt = (L + 31) >> 5;
    int wave = blockIdx.x * (blockDim.x >> 5) + (threadIdx.x >> 5);
    if (wave >= Mt * Nt) return;            // whole-wave exit; EXEC stays full
    int tm = wave / Nt, tn = wave - tm * Nt;
    int lane = threadIdx.x & 31;
    int half = lane >> 4, lr = lane & 15;
    int h2 = half * 2;
    int mA0 = tm * 32 + lr,      mA1 = tm * 32 + 16 + lr;
    int nB0 = tn * 32 + lr,      nB1 = tn * 32 + 16 + lr;
    bool nOk0 = nB0 < Cout,      nOk1 = nB1 < Cout;

    v8f acc00, acc01, acc10, acc11;
#pragma unroll
    for (int j = 0; j < 8; ++j) { acc00[j] = 0.f; acc01[j] = 0.f; acc10[j] = 0.f; acc11[j] = 0.f; }

    if ((Cin & 3) == 0) {
        // ---- division-free path: per-tap GEMM accumulation ----
        for (int t = 0; t < ks; ++t) {
            int sl0 = mA0 + t - pad;
            int sl1 = mA1 + t - pad;
            bool okA0 = (mA0 < L) && (sl0 >= 0) && (sl0 < L);
            bool okA1 = (mA1 < L) && (sl1 >= 0) && (sl1 < L);
            const float* Xr0 = X + (size_t)sl0 * Cin;
            const float* Xr1 = X + (size_t)sl1 * Cin;
            const float* Wt  = W + (size_t)t * Cin * Cout;
#pragma unroll 2
            for (int ci0 = 0; ci0 < Cin; ci0 += 4) {
                v2f a0, a1, b0, b1;
#pragma unroll
                for (int j = 0; j < 2; ++j) {
                    int ci = ci0 + h2 + j;
                    float x0 = okA0 ? Xr0[ci] : 0.f;
                    float x1 = okA1 ? Xr1[ci] : 0.f;
                    if (bng) {
                        float sc = bng[ci] * rsqrtf(bnv[ci] + 1e-3f);
                        float of = bnb[ci] - sc * bnm[ci];
                        x0 = okA0 ? gelu_(x0 * sc + of) : 0.f;
                        x1 = okA1 ? gelu_(x1 * sc + of) : 0.f;
                    }
                    a0[j] = x0; a1[j] = x1;
                    const float* Wr = Wt + (size_t)ci * Cout;
                    b0[j] = nOk0 ? Wr[nB0] : 0.f;
                    b1[j] = nOk1 ? Wr[nB1] : 0.f;
                }
                acc00 = wmma4_(a0, b0, acc00);
                acc01 = wmma4_(a0, b1, acc01);
                acc10 = wmma4_(a1, b0, acc10);
                acc11 = wmma4_(a1, b1, acc11);
            }
        }
    } else {
        // ---- generic path (used only for Cin==1 seq stem, K=ks) ----
        int K = ks * Cin;
        for (int k0 = 0; k0 < K; k0 += 4) {
            v2f a0, a1, b0, b1;
#pragma unroll
            for (int j = 0; j < 2; ++j) {
                int kk = k0 + h2 + j;
                float x0 = 0.f, x1 = 0.f, w0 = 0.f, w1 = 0.f;
                if (kk < K) {
                    int t = kk;                         // Cin == 1 -> ci = 0
                    int sl0 = mA0 + t - pad;
                    int sl1 = mA1 + t - pad;
                    if (mA0 < L && sl0 >= 0 && sl0 < L) {
                        x0 = X[sl0];
                        if (bng) x0 = gelu_(bng[0] * (x0 - bnm[0]) * rsqrtf(bnv[0] + 1e-3f) + bnb[0]);
                    }
                    if (mA1 < L && sl1 >= 0 && sl1 < L) {
                        x1 = X[sl1];
                        if (bng) x1 = gelu_(bng[0] * (x1 - bnm[0]) * rsqrtf(bnv[0] + 1e-3f) + bnb[0]);
                    }
                    if (nOk0) w0 = W[(size_t)kk * Cout + nB0];
                    if (nOk1) w1 = W[(size_t)kk * Cout + nB1];
                }
                a0[j] = x0; a1[j] = x1; b0[j] = w0; b1[j] = w1;
            }
            acc00 = wmma4_(a0, b0, acc00);
            acc01 = wmma4_(a0, b1, acc01);
            acc10 = wmma4_(a1, b0, acc10);
            acc11 = wmma4_(a1, b1, acc11);
        }
    }

    // ---- epilogue: bias / residual / activation ----
    const v8f* accs[4] = { &acc00, &acc01, &acc10, &acc11 };
#pragma unroll
    for (int i = 0; i < 2; ++i) {
#pragma unroll
        for (int jn = 0; jn < 2; ++jn) {
            int n = tn * 32 + jn * 16 + lr;
            if (n >= Cout) continue;
            float bs = bias ? bias[n] : 0.f;
            const v8f& a = *accs[i * 2 + jn];
#pragma unroll
            for (int j = 0; j < 8; ++j) {
                int mm = tm * 32 + i * 16 + half * 8 + j;   // C/D: M = vgpr + half*8
                if (mm < L) {
                    float v = a[j] + bs;
                    if (res) v += res[(size_t)mm * ldc + n];
                    if (act == 1)      v = gelu_(v);
                    else if (act == 2) v = fmaxf(v, 0.f);
                    else if (act == 3) v = 1.f / (1.f + expf(-v));
                    C[(size_t)mm * ldc + n] = v;
                }
            }
        }
    }
}

// ======================= batched WMMA GEMM (attention) =====================
__global__ __launch_bounds__(128)
void k_bgemm(
    const float* __restrict__ A, int lda, long long sA,
    const float* __restrict__ B, int ldb, long long sB, int transB,
    float* __restrict__ Cc, int ldc, long long sC,
    int M, int N, int K)
{
    A  += (long long)blockIdx.z * sA;
    B  += (long long)blockIdx.z * sB;
    Cc += (long long)blockIdx.z * sC;
    int Nt = (N + 31) >> 5, Mt = (M + 31) >> 5;
    int wave = blockIdx.x * (blockDim.x >> 5) + (threadIdx.x >> 5);
    if (wave >= Mt * Nt) return;
    int tm = wave / Nt, tn = wave - tm * Nt;
    int lane = threadIdx.x & 31;
    int half = lane >> 4, lr = lane & 15;
    int h2 = half * 2;
    int mA0 = tm * 32 + lr,  mA1 = tm * 32 + 16 + lr;
    int nB0 = tn * 32 + lr,  nB1 = tn * 32 + 16 + lr;
    bool mOk0 = mA0 < M, mOk1 = mA1 < M;
    bool nOk0 = nB0 < N, nOk1 = nB1 < N;

    v8f acc00, acc01, acc10, acc11;
#pragma unroll
    for (int j = 0; j < 8; ++j) { acc00[j] = 0.f; acc01[j] = 0.f; acc10[j] = 0.f; acc11[j] = 0.f; }

    const float* Ar0 = A + (size_t)mA0 * lda;
    const float* Ar1 = A + (size_t)mA1 * lda;
    const float* Bt0 = transB ? (B + (size_t)nB0 * ldb) : (B + nB0);
    const float* Bt1 = transB ? (B + (size_t)nB1 * ldb) : (B + nB1);

#pragma unroll 2
    for (int k0 = 0; k0 < K; k0 += 4) {             // all call sites: K % 4 == 0
        v2f a0, a1, b0, b1;
#pragma unroll
        for (int j = 0; j < 2; ++j) {
            int kk = k0 + h2 + j;
            a0[j] = mOk0 ? Ar0[kk] : 0.f;
            a1[j] = mOk1 ? Ar1[kk] : 0.f;
            if (transB) {
                b0[j] = nOk0 ? Bt0[kk] : 0.f;
                b1[j] = nOk1 ? Bt1[kk] : 0.f;
            } else {
                b0[j] = nOk0 ? Bt0[(size_t)kk * ldb] : 0.f;
                b1[j] = nOk1 ? Bt1[(size_t)kk * ldb] : 0.f;
            }
        }
        acc00 = wmma4_(a0, b0, acc00);
        acc01 = wmma4_(a0, b1, acc01);
        acc10 = wmma4_(a1, b0, acc10);
        acc11 = wmma4_(a1, b1, acc11);
    }

    const v8f* accs[4] = { &acc00, &acc01, &acc10, &acc11 };
#pragma unroll
    for (int i = 0; i < 2; ++i) {
#pragma unroll
        for (int jn = 0; jn < 2; ++jn) {
            int n = tn * 32 + jn * 16 + lr;
            if (n >= N) continue;
            const v8f& a = *accs[i * 2 + jn];
#pragma unroll
            for (int j = 0; j < 8; ++j) {
                int mm = tm * 32 + i * 16 + half * 8 + j;
                if (mm < M) Cc[(size_t)mm * ldc + n] = a[j];
            }
        }
    }
}

// ============================ elementwise kernels ==========================
__global__ void k_pool(const float* __restrict__ in, int Lout, int C,
                       float* __restrict__ out, int ldo)
{
    int idx = blockIdx.x * blockDim.x + threadIdx.x;
    if (idx >= Lout * C) return;
    int l = idx / C, c = idx - l * C;
    float a = in[(size_t)(2 * l) * C + c];
    float b = in[(size_t)(2 * l + 1) * C + c];
    out[(size_t)l * ldo + c] = fmaxf(a, b);
}

__global__ void k_ln(const float* __restrict__ x, const float* __restrict__ g,
                     const float* __restrict__ b, float* __restrict__ y, int C)
{
    int row = blockIdx.x;
    const float* xr = x + (size_t)row * C;
    float s = 0.f, s2 = 0.f;
    for (int i = threadIdx.x; i < C; i += 32) { float v = xr[i]; s += v; s2 += v * v; }
    for (int o = 16; o > 0; o >>= 1) { s += __shfl_xor(s, o, 32); s2 += __shfl_xor(s2, o, 32); }
    float mu = s / C;
    float var = s2 / C - mu * mu;
    float inv = rsqrtf(var + 1e-3f);
    float* yr = y + (size_t)row * C;
    for (int i = threadIdx.x; i < C; i += 32)
        yr[i] = g[i] * (xr[i] - mu) * inv + b[i];
}

__global__ void k_qrwrr(const float* __restrict__ q, const float* __restrict__ rw,
                        const float* __restrict__ rr, float* __restrict__ qrw,
                        float* __restrict__ qrr)
{
    int idx = blockIdx.x * blockDim.x + threadIdx.x;
    if (idx >= 8 * 512 * 64) return;
    int d = idx & 63;
    int l = (idx >> 6) & 511;
    int h = idx >> 15;
    float qv = q[(size_t)l * 512 + h * 64 + d] * 0.125f;   // KS^-0.5
    qrw[idx] = qv + rw[h * 64 + d];
    qrr[idx] = qv + rr[h * 64 + d];
}

// softmax over (content + relative_shift(rel)) rows; in-place on cont
__global__ void k_softmax_shift(float* __restrict__ cont, const float* __restrict__ rel)
{
    int row = blockIdx.x;          // h*512 + l
    int h = row >> 9, l = row & 511;
    float* crow = cont + (size_t)row * 512;
    const float* rb = rel + (size_t)h * 512 * 1023;
    int lane = threadIdx.x;
    float v[16];
    float mx = -3.4e38f;
#pragma unroll
    for (int i = 0; i < 16; ++i) {
        int m = lane + i * 32;
        int f = 512 + l * 1023 + m;            // flat index into padded/reshaped
        int j = f & 1023;
        int sl = f >> 10;
        float sh = (j == 0) ? 0.f : rb[(size_t)sl * 1023 + (j - 1)];
        float val = crow[m] + sh;
        v[i] = val;
        mx = fmaxf(mx, val);
    }
#pragma unroll
    for (int o = 16; o > 0; o >>= 1) mx = fmaxf(mx, __shfl_xor(mx, o, 32));
    float sum = 0.f;
#pragma unroll
    for (int i = 0; i < 16; ++i) { v[i] = expf(v[i] - mx); sum += v[i]; }
#pragma unroll
    for (int o = 16; o > 0; o >>= 1) sum += __shfl_xor(sum, o, 32);
    float inv = 1.f / sum;
#pragma unroll
    for (int i = 0; i < 16; ++i) crow[lane + i * 32] = v[i] * inv;
}

// ====================== positional features (L = 512) ======================
DEV float gamma_prob_(float ap, int i)
{
    float mean = 64.f + 448.f * (float)i / 7.f;      // linspace(64, 512, 8)
    float c    = (mean / 32.f) * (mean / 32.f);      // conc
    float rate = mean / 1024.f;                      // mean / std^2
    float log_un = ((ap == 0.f) ? 0.f
                                : (c - 1.f) * logf(fmaxf(ap, 1e-20f))) - rate * ap;
    float log_nm = lgammaf(c) - c * logf(rate);
    return expf(log_un - log_nm) + 1e-8f;
}

__global__ void k_prob(float* __restrict__ prob)
{
    int idx = blockIdx.x * blockDim.x + threadIdx.x;
    if (idx >= 1023 * 8) return;
    int r = idx >> 3, i = idx & 7;
    float ap = fabsf((float)(r - 511));
    prob[idx] = gamma_prob_(ap, i);
}

__global__ void k_gmax(const float* __restrict__ p, int n, float* __restrict__ out)
{
    __shared__ float sm[256];
    float mx = -3.4e38f;
    for (int i = threadIdx.x; i < n; i += 256) mx = fmaxf(mx, p[i]);
    sm[threadIdx.x] = mx;
    __syncthreads();
    for (int s = 128; s > 0; s >>= 1) {
        if ((int)threadIdx.x < s) sm[threadIdx.x] = fmaxf(sm[threadIdx.x], sm[threadIdx.x + s]);
        __syncthreads();
    }
    if (threadIdx.x == 0) *out = sm[0];
}

__global__ void k_pe(float* __restrict__ pe, const float* __restrict__ pmax)
{
    int r = blockIdx.x * blockDim.x + threadIdx.x;
    if (r >= 1023) return;
    float pos = (float)(r - 511);
    float ap  = fabsf(pos);
    float sgn = (pos > 0.f) ? 1.f : ((pos < 0.f) ? -1.f : 0.f);
    float mx = *pmax;
    float* row = pe + (size_t)r * 48;
    for (int i = 0; i < 8; ++i) {
        float e  = 3.f + 6.f * (float)i / 7.f;       // linspace(3, log2(512), 8)
        float hl = exp2f(e);
        float fe = expf(-0.69314718056f / hl * ap);
        float cw = exp2f((float)(i + 1)) - 1.f;
        float fc = (cw > ap) ? 1.f : 0.f;
        float fg = gamma_prob_(ap, i) / mx;
        row[i]      = fe; row[8 + i]  = fc; row[16 + i] = fg;
        row[24 + i] = sgn * fe; row[32 + i] = sgn * fc; row[40 + i] = sgn * fg;
    }
}

// ============================ parameter plumbing ===========================
struct BNP { const float *b, *g, *m, *v; };
struct CVP { const float *b, *w; };
struct CBP { BNP bn; CVP conv; };
struct TrunkP { CVP stem; CBP stem_res; CBP cb[6]; CBP rb[6]; };
struct AttnP { const float *bo, *rr, *rw, *wk, *wo, *wq, *wr, *wv; };
struct LayerP { AttnP attn; const float *b1, *b2, *ln1b, *ln1g, *ln2b, *ln2g, *w1, *w2; };
struct HeadP { CVP din, dmid, dout; CBP fp[3]; LayerP ly[2]; };

struct Cur { void* const* d; int i; const float* nx() { return (const float*)d[i++]; } };

static void pBN(Cur& c, BNP& p) { p.b = c.nx(); p.g = c.nx(); p.m = c.nx(); p.v = c.nx(); }
static void pCV(Cur& c, CVP& p) { p.b = c.nx(); p.w = c.nx(); }
static void pCB(Cur& c, CBP& p) { pBN(c, p.bn); pCV(c, p.conv); }
static void pTrunk(Cur& c, TrunkP& t) {
    pCV(c, t.stem); pCB(c, t.stem_res);
    for (int i = 0; i < 6; ++i) { pCB(c, t.cb[i]); pCB(c, t.rb[i]); }
}
static void pHead(Cur& c, HeadP& h) {
    pCV(c, h.din); pCV(c, h.dmid); pCV(c, h.dout);
    for (int i = 0; i < 3; ++i) pCB(c, h.fp[i]);
    for (int l = 0; l < 2; ++l) {
        LayerP& L = h.ly[l];
        AttnP& a = L.attn;
        a.bo = c.nx(); a.rr = c.nx(); a.rw = c.nx(); a.wk = c.nx();
        a.wo = c.nx(); a.wq = c.nx(); a.wr = c.nx(); a.wv = c.nx();
        L.b1 = c.nx(); L.b2 = c.nx();
        L.ln1b = c.nx(); L.ln1g = c.nx();
        L.ln2b = c.nx(); L.ln2g = c.nx();
        L.w1 = c.nx(); L.w2 = c.nx();
    }
}

// ============================ launch helpers ===============================
static inline void gemm(hipStream_t s, const float* X, int L, int Cin,
                        const float* W, int Cout, const float* bias,
                        const float* res, float* C, int ldc,
                        int ks, int pad, int act, const BNP* bn)
{
    int tiles = ((L + 31) / 32) * ((Cout + 31) / 32);
    int blocks = (tiles + 3) / 4;
    k_conv_wmma<<<blocks, 128, 0, s>>>(X, L, Cin, W, Cout, bias, res, C, ldc,
        ks, pad, act,
        bn ? bn->g : nullptr, bn ? bn->b : nullptr,
        bn ? bn->m : nullptr, bn ? bn->v : nullptr);
}

static inline void bgemm(hipStream_t s, const float* A, int lda, long long sA,
                         const float* B, int ldb, long long sB, int transB,
                         float* C, int ldc, long long sC,
                         int M, int N, int K, int batch)
{
    int tiles = ((M + 31) / 32) * ((N + 31) / 32);
    dim3 g((tiles + 3) / 4, 1, batch);
    k_bgemm<<<g, 128, 0, s>>>(A, lda, sA, B, ldb, sB, transB, C, ldc, sC, M, N, K);
}

static inline void pool(hipStream_t s, const float* in, int Lout, int C,
                        float* out, int ldo)
{
    int n = Lout * C;
    k_pool<<<(n + 255) / 256, 256, 0, s>>>(in, Lout, C, out, ldo);
}

// ============================ workspace layout (floats) ====================
static const size_t BIG0  = 0;
static const size_t BIG1  = 12582912;            // 65536*192
static const size_t XCAT  = 25165824;            // 512*768
static const size_t PE    = XCAT + 393216;       // 1023*48
static const size_t PROB  = PE + 49104;          // 1023*8
static const size_t PMAX  = PROB + 8184;         // 1
// per-head scratch (reuses the big region, trunks already done)
static const size_t OHX   = 0;                   // 512*384
static const size_t OHXN  = 196608;              // 512*384
static const size_t OHQ   = 393216;              // 512*512
static const size_t OHK   = 655360;              // 512*512
static const size_t OHV   = 917504;              // 512*384
static const size_t OHQRW = 1114112;             // 8*512*64
static const size_t OHQRR = 1376256;             // 8*512*64
static const size_t OHRK  = 1638400;             // 1023*512
static const size_t OHCONT= 2162176;             // 8*512*512
static const size_t OHREL = 4259328;             // 8*512*1023
static const size_t OHO   = 8449536;             // 512*384
static const size_t OHFF  = 8646144;             // 512*768
static const size_t OHT1  = 9039360;             // 448*768
static const size_t OHT2  = 9383424;             // 448*1536

// ============================ model stages =================================
static void run_trunk(hipStream_t s, const TrunkP& tp, const float* input, int cin,
                      float* buf0, float* buf1, float* xcat_out)
{
    static const int F_[6] = {192, 192, 256, 320, 320, 384};
    int L = 65536;
    // stem: conv15 cin->192
    gemm(s, input, L, cin, tp.stem.w, 192, tp.stem.b, nullptr, buf0, 192, 15, 7, 0, nullptr);
    // x += conv1(gelu(bn(x)))   (BN+GELU fused into A-gather)
    gemm(s, buf0, L, 192, tp.stem_res.conv.w, 192, tp.stem_res.conv.b, buf0,
         buf1, 192, 1, 0, 0, &tp.stem_res.bn);
    pool(s, buf1, L / 2, 192, buf0, 192);
    L /= 2;
    float* P = buf0; float* Q = buf1;
    int ci = 192;
    for (int bI = 0; bI < 6; ++bI) {
        int co = F_[bI];
        // cb: conv5(gelu(bn(P)))
        gemm(s, P, L, ci, tp.cb[bI].conv.w, co, tp.cb[bI].conv.b, nullptr,
             Q, co, 5, 2, 0, &tp.cb[bI].bn);
        // rb: Q + conv1(gelu(bn(Q)))
        gemm(s, Q, L, co, tp.rb[bI].conv.w, co, tp.rb[bI].conv.b, Q,
             P, co, 1, 0, 0, &tp.rb[bI].bn);
        if (bI == 5) {
            pool(s, P, L / 2, co, xcat_out, 768);    // write into concat slot
        } else {
            pool(s, P, L / 2, co, Q, co);
            float* t = P; P = Q; Q = t;
        }
        L /= 2;
        ci = co;
    }
}

static void run_head(hipStream_t s, const HeadP& h, float* ws, float* out,
                     int units, int colOff)
{
    const float* xcat = ws + XCAT;
    const float* pe   = ws + PE;
    float* HX   = ws + OHX;
    float* HXN  = ws + OHXN;
    float* HQ   = ws + OHQ;
    float* HK   = ws + OHK;
    float* HV   = ws + OHV;
    float* HQRW = ws + OHQRW;
    float* HQRR = ws + OHQRR;
    float* HRK  = ws + OHRK;
    float* HCONT= ws + OHCONT;
    float* HREL = ws + OHREL;
    float* HO   = ws + OHO;
    float* HFF  = ws + OHFF;
    float* HT1  = ws + OHT1;
    float* HT2  = ws + OHT2;

    // din: (512,768) @ (768,384)
    gemm(s, xcat, 512, 768, h.din.w, 384, h.din.b, nullptr, HX, 384, 1, 0, 0, nullptr);

    for (int l = 0; l < 2; ++l) {
        const LayerP& L = h.ly[l];
        k_ln<<<512, 32, 0, s>>>(HX, L.ln1g, L.ln1b, HXN, 384);
        gemm(s, HXN, 512, 384, L.attn.wq, 512, nullptr, nullptr, HQ, 512, 1, 0, 0, nullptr);
        gemm(s, HXN, 512, 384, L.attn.wk, 512, nullptr, nullptr, HK, 512, 1, 0, 0, nullptr);
        gemm(s, HXN, 512, 384, L.attn.wv, 384, nullptr, nullptr, HV, 384, 1, 0, 0, nullptr);
        k_qrwrr<<<(8 * 512 * 64 + 255) / 256, 256, 0, s>>>(HQ, L.attn.rw, L.attn.rr, HQRW, HQRR);
        // rk = pe @ wr : (1023,48)@(48,512)
        gemm(s, pe, 1023, 48, L.attn.wr, 512, nullptr, nullptr, HRK, 512, 1, 0, 0, nullptr);
        // content[h] = qrw[h] @ k[h]^T
        bgemm(s, HQRW, 64, 512 * 64, HK, 512, 64, 1, HCONT, 512, 512 * 512,
              512, 512, 64, 8);
        // rel[h] = qrr[h] @ rk[h]^T
        bgemm(s, HQRR, 64, 512 * 64, HRK, 512, 64, 1, HREL, 1023, 512 * 1023,
              512, 1023, 64, 8);
        k_softmax_shift<<<8 * 512, 32, 0, s>>>(HCONT, HREL);
        // o[h] = a[h] @ v[h]  -> interleaved (512, H*48)
        bgemm(s, HCONT, 512, 512 * 512, HV, 384, 48, 0, HO, 384, 48,
              512, 48, 512, 8);
        // x = x + o @ wo + bo
        gemm(s, HO, 512, 384, L.attn.wo, 384, L.attn.bo, HX, HX, 384, 1, 0, 0, nullptr);
        // FFN
        k_ln<<<512, 32, 0, s>>>(HX, L.ln2g, L.ln2b, HXN, 384);
        gemm(s, HXN, 512, 384, L.w1, 768, L.b1, nullptr, HFF, 768, 1, 0, 2, nullptr);
        gemm(s, HFF, 512, 768, L.w2, 384, L.b2, HX, HX, 384, 1, 0, 0, nullptr);
    }

    // trim rows [32, 480) then dmid
    gemm(s, HX + 32 * 384, 448, 384, h.dmid.w, 384, h.dmid.b, nullptr,
         HXN, 384, 1, 0, 0, nullptr);
    // fp blocks: x = gelu(conv1(gelu(bn(x))) + b)
    gemm(s, HXN, 448, 384,  h.fp[0].conv.w, 768,  h.fp[0].conv.b, nullptr,
         HT1, 768, 1, 0, 1, &h.fp[0].bn);
    gemm(s, HT1, 448, 768,  h.fp[1].conv.w, 1536, h.fp[1].conv.b, nullptr,
         HT2, 1536, 1, 0, 1, &h.fp[1].bn);
    gemm(s, HT2, 448, 1536, h.fp[2].conv.w, 768,  h.fp[2].conv.b, nullptr,
         HT1, 768, 1, 0, 1, &h.fp[2].bn);
    // dout with fused sigmoid, written straight into the concat output slice
    gemm(s, HT1, 448, 768, h.dout.w, units, h.dout.b, nullptr,
         out + colOff, 90, 1, 0, 3, nullptr);
}

// ================================ entry ====================================
extern "C" void kernel_launch(void* const* d_in, const int* in_sizes, int n_in,
                              void* d_out, int out_size, void* d_ws, size_t ws_size,
                              hipStream_t stream)
{
    (void)in_sizes; (void)n_in; (void)out_size; (void)ws_size;
    const float* dna = (const float*)d_in[0];   // (65536, 4)
    const float* seq = (const float*)d_in[1];   // (65536, 1)

    Cur c{d_in, 2};
    TrunkP tdna, tseq;
    pTrunk(c, tdna);
    pTrunk(c, tseq);
    // flatten order of head_* keys is string-sorted: 0,1,10,2,...,9
    static const int order[11] = {0, 1, 10, 2, 3, 4, 5, 6, 7, 8, 9};
    HeadP heads[11];
    for (int k = 0; k < 11; ++k) pHead(c, heads[order[k]]);
    // trailing is_training leaf ignored

    float* ws = (float*)d_ws;
    float* out = (float*)d_out;

    // trunks -> xcat (512, 768)
    run_trunk(stream, tdna, dna, 4, ws + BIG0, ws + BIG1, ws + XCAT + 0);
    run_trunk(stream, tseq, seq, 1, ws + BIG0, ws + BIG1, ws + XCAT + 384);

    // positional features (1023, 48)
    k_prob<<<(1023 * 8 + 127) / 128, 128, 0, stream>>>(ws + PROB);
    k_gmax<<<1, 256, 0, stream>>>(ws + PROB, 1023 * 8, ws + PMAX);
    k_pe<<<(1023 + 127) / 128, 128, 0, stream>>>(ws + PE, ws + PMAX);

    static const int units[11] = {73, 8, 1, 1, 1, 1, 1, 1, 1, 1, 1};
    int off = 0;
    for (int i = 0; i < 11; ++i) {
        run_head(stream, heads[i], ws, out, units[i], off);
        off += units[i];
    }
}